// RGCNSegmentMM_3908420239949
// MI455X (gfx1250) — compile-verified
//
#include <hip/hip_runtime.h>
#include <stdint.h>

#define N_NODES 100000
#define N_EDGES 1600000
#define NUM_RELS 8
#define IN_FEAT 128
#define OUT_FEAT 128

typedef __attribute__((ext_vector_type(2))) float v2f;
typedef __attribute__((ext_vector_type(8))) float v8f;

// 16-byte payload type for the async-to-LDS builtin (matches the builtin's
// declared parameter type: 'int __vector(4)' pointers in AS1 / AS3).
typedef int b128_t __attribute__((vector_size(16)));
typedef __attribute__((address_space(1))) b128_t* global_b128_ptr;
typedef __attribute__((address_space(3))) b128_t* lds_b128_ptr;

// LDS row strides padded to avoid bank conflicts (64 banks x 4B).
#define A_STRIDE 132
#define W_STRIDE 132

// ---------------------------------------------------------------------------
// CDNA5 async direct-to-LDS copy (ASYNCcnt-tracked), 16B per lane.
// ---------------------------------------------------------------------------
__device__ __forceinline__ void async_copy_b128(const float* gsrc, float* ldst) {
#if __has_builtin(__builtin_amdgcn_global_load_async_to_lds_b128)
    __builtin_amdgcn_global_load_async_to_lds_b128(
        (global_b128_ptr)gsrc,
        (lds_b128_ptr)ldst,
        /*offset=*/0, /*cpol=*/0);
#else
    const uint32_t lds_addr = (uint32_t)(uintptr_t)ldst;  // LDS aperture low bits
    asm volatile("global_load_async_to_lds_b128 %0, %1, off"
                 :
                 : "v"(lds_addr), "v"(gsrc)
                 : "memory");
#endif
}

__device__ __forceinline__ void wait_asynccnt0() {
#if __has_builtin(__builtin_amdgcn_s_wait_asynccnt)
    __builtin_amdgcn_s_wait_asynccnt(0);
#else
    asm volatile("s_wait_asynccnt 0x0" ::: "memory");
#endif
}

// ---------------------------------------------------------------------------
// GEMM: proj[m, n] = sum_k feat[m, k] * W[rel][k, n]
// Block: 128 threads = 4 waves (wave32). Each block computes a 128(M) x 64(N)
// strip. Weights (64 n x 128 k, transposed) staged once in LDS; feat staged
// 16 rows at a time via double-buffered GLOBAL_LOAD_ASYNC_TO_LDS_B128.
// Each wave owns one 16x16 tile: 32 x V_WMMA_F32_16X16X4_F32 over K=128.
//
// WMMA f32 16x16x4 VGPR layout (ISA 7.12.2):
//   A (16x4 MxK):  lanes 0-15: M=lane, v0=K0, v1=K1 ; lanes 16-31: K2,K3
//   B (4x16 KxN):  lanes 0-15: N=lane, v0=K0, v1=K1 ; lanes 16-31: K2,K3
//   C/D (16x16):   VGPR v: lanes 0-15 -> M=v, N=lane; lanes 16-31 -> M=v+8
// ---------------------------------------------------------------------------
__global__ __launch_bounds__(128) void rgcn_gemm_kernel(
    const float* __restrict__ feat, const float* __restrict__ weight,
    float* __restrict__ proj, int rel)
{
    __shared__ float Wt[64 * W_STRIDE];      // Wt[n_local][k]
    __shared__ float At[2][16 * A_STRIDE];   // double-buffered At[m_local][k]

    const int t    = threadIdx.x;
    const int lane = t & 31;
    const int wave = t >> 5;
    const int half = lane >> 4;   // 0: lanes 0-15, 1: lanes 16-31
    const int lrow = lane & 15;

    const int n_base  = blockIdx.y * 64;
    const int m_block = blockIdx.x * 128;
    const float* W = weight + rel * (IN_FEAT * OUT_FEAT);  // W[k][n]

    // Stage 64(n) x 128(k) weight chunk into LDS, transposed to Wt[n][k].
    for (int j = 0; j < 64; ++j) {
        const int f = j * 128 + t;
        const int k = f >> 6;
        const int n = f & 63;
        Wt[n * W_STRIDE + k] = W[k * OUT_FEAT + n_base + n];
    }

    // Prologue: async-prefetch feat subtile 0 into At[0].
    // 16 rows x 128 k = 512 x b128 across 128 threads (4 per thread).
    {
        for (int j = 0; j < 4; ++j) {
            const int f4  = j * 128 + t;
            const int row = f4 >> 5;
            const int kq  = (f4 & 31) * 4;
            async_copy_b128(feat + (m_block + row) * IN_FEAT + kq,
                            &At[0][row * A_STRIDE + kq]);
        }
    }

    const int n0 = n_base + wave * 16;  // this wave's 16 output columns

    for (int ms = 0; ms < 8; ++ms) {
        const int m_sub = m_block + ms * 16;

        wait_asynccnt0();   // my async writes into At[ms&1] landed
        __syncthreads();    // all waves' writes visible; all done reading
                            // At[(ms+1)&1] (last read in iteration ms-1)

        // Prefetch next subtile into the other buffer (block-uniform branch).
        const int m_next = m_block + (ms + 1) * 16;
        if (ms + 1 < 8 && m_next < N_NODES) {
            float* buf = At[(ms + 1) & 1];
            for (int j = 0; j < 4; ++j) {
                const int f4  = j * 128 + t;
                const int row = f4 >> 5;
                const int kq  = (f4 & 31) * 4;
                async_copy_b128(feat + (m_next + row) * IN_FEAT + kq,
                                &buf[row * A_STRIDE + kq]);
            }
        }

        if (m_sub < N_NODES) {
            const float* Ab = At[ms & 1];
            v8f c = {};
            #pragma unroll
            for (int kk = 0; kk < 32; ++kk) {
                const int k = kk * 4;
                const v2f a = *(const v2f*)(Ab + lrow * A_STRIDE + k + half * 2);
                const v2f b = *(const v2f*)(Wt + (wave * 16 + lrow) * W_STRIDE
                                            + k + half * 2);
                c = __builtin_amdgcn_wmma_f32_16x16x4_f32(
                        /*neg_a=*/false, a, /*neg_b=*/false, b,
                        /*c_mod=*/(short)0, c,
                        /*reuse_a=*/false, /*reuse_b=*/false);
            }
            // Write D: lane holds rows m_sub + v + half*8, column n0 + lrow.
            float* outp = proj + (m_sub + half * 8) * OUT_FEAT + n0 + lrow;
            #pragma unroll
            for (int v = 0; v < 8; ++v) {
                outp[v * OUT_FEAT] = c[v];
            }
        }
    }
}

// ---------------------------------------------------------------------------
// Scatter: for each edge with etypes[e] == rel:
//   out[dst[e], :] += proj[src[e], :]
// One wave per edge; each lane moves a float4 (32 x 4 = 128 features).
// proj (51.2 MB) and out (51.2 MB) both fit in the 192 MB L2, so the random
// gathers and the fp32 atomics stay on-chip.
// ---------------------------------------------------------------------------
__global__ __launch_bounds__(256) void rgcn_scatter_kernel(
    const float* __restrict__ proj, const int* __restrict__ src,
    const int* __restrict__ dst, const int* __restrict__ etypes,
    int rel, float* __restrict__ out)
{
    const int e    = (int)((blockIdx.x * 256u + threadIdx.x) >> 5);
    const int lane = threadIdx.x & 31;
    if (e >= N_EDGES) return;
    if (etypes[e] != rel) return;  // uniform per wave (all lanes same e)

    const int s = src[e];
    const int d = dst[e];
    const float4 v = *(const float4*)(proj + s * OUT_FEAT + lane * 4);
    float* o = out + d * OUT_FEAT + lane * 4;
    atomicAdd(o + 0, v.x);
    atomicAdd(o + 1, v.y);
    atomicAdd(o + 2, v.z);
    atomicAdd(o + 3, v.w);
}

__global__ __launch_bounds__(256) void rgcn_zero_kernel(float4* __restrict__ out,
                                                        int n4)
{
    const int i = blockIdx.x * 256 + threadIdx.x;
    if (i < n4) out[i] = float4{0.f, 0.f, 0.f, 0.f};
}

extern "C" void kernel_launch(void* const* d_in, const int* in_sizes, int n_in,
                              void* d_out, int out_size, void* d_ws, size_t ws_size,
                              hipStream_t stream) {
    const float* feat   = (const float*)d_in[0];  // [100000, 128] f32
    const float* weight = (const float*)d_in[1];  // [8, 128, 128] f32
    const int*   src    = (const int*)d_in[2];    // [1.6M] i32
    const int*   dst    = (const int*)d_in[3];    // [1.6M] i32
    const int*   etyp   = (const int*)d_in[4];    // [1.6M] i32
    float* out  = (float*)d_out;                  // [100000, 128] f32
    float* proj = (float*)d_ws;                   // 100000*128 f32 = 51.2 MB

    // out is poisoned by the harness; zero it before accumulation.
    rgcn_zero_kernel<<<(N_NODES * OUT_FEAT / 4 + 255) / 256, 256, 0, stream>>>(
        (float4*)out, N_NODES * OUT_FEAT / 4);

    const dim3 ggrid((N_NODES + 127) / 128, 2);
    const int sblocks = (N_EDGES * 32 + 255) / 256;  // one wave per edge
    for (int r = 0; r < NUM_RELS; ++r) {
        // Same-stream ordering guarantees GEMM_r completes before scatter_r.
        rgcn_gemm_kernel<<<ggrid, 128, 0, stream>>>(feat, weight, proj, r);
        rgcn_scatter_kernel<<<sblocks, 256, 0, stream>>>(proj, src, dst, etyp,
                                                         r, out);
    }
}